// MSELossWithPenalty_14508399526012
// MI455X (gfx1250) — compile-verified
//
#include <hip/hip_runtime.h>
#include <hip/hip_bf16.h>

typedef __attribute__((ext_vector_type(2))) float v2f;
typedef __attribute__((ext_vector_type(8))) float v8f;

#define NBLK 2048
#define NTHR 256
#define WAVES_PER_BLK (NTHR / 32)

// Exact fp32 sum across the 32 lanes of a wave, broadcast to all lanes.
// Uses V_WMMA_F32_16X16X4_F32 twice:
//   Stage 1: A = {x, 0} per lane -> per documented 16x4 A layout,
//            row m = {x[m], 0, x[m+16], 0}; B = ones => D[m][n] = x[m]+x[m+16].
//            Summing D's 8 VGPRs in-lane gives S_lo (lanes 0-15) / S_hi (16-31).
//   Stage 2: A = {S, 0} -> D2[m][n] = S_lo + S_hi = full wave sum, in all lanes.
__device__ __forceinline__ float wave_sum_f32(float x) {
#if __has_builtin(__builtin_amdgcn_wmma_f32_16x16x4_f32)
  v2f ones; ones.x = 1.0f; ones.y = 1.0f;
  v8f c = {};
  v2f a; a.x = x; a.y = 0.0f;
  v8f d = __builtin_amdgcn_wmma_f32_16x16x4_f32(
      /*neg_a=*/false, a, /*neg_b=*/false, ones,
      /*c_mod=*/(short)0, c, /*reuse_a=*/false, /*reuse_b=*/false);
  float half_sum = ((d[0] + d[1]) + (d[2] + d[3])) +
                   ((d[4] + d[5]) + (d[6] + d[7]));
  v2f a2; a2.x = half_sum; a2.y = 0.0f;
  v8f d2 = __builtin_amdgcn_wmma_f32_16x16x4_f32(
      false, a2, false, ones, (short)0, c, false, false);
  return d2[0];
#else
  for (int off = 16; off > 0; off >>= 1)
    x += __shfl_xor(x, off, 32);
  return x;
#endif
}

__device__ __forceinline__ float sign_mismatch(float p, float t) {
  // jnp.sign semantics: {-1, 0, +1}; mismatch iff sign classes differ.
  bool m = ((p > 0.0f) != (t > 0.0f)) || ((p < 0.0f) != (t < 0.0f));
  return m ? 1.0f : 0.0f;
}

__global__ void mse_penalty_partial_kernel(const float* __restrict__ pred,
                                           const float* __restrict__ targ,
                                           int n, float2* __restrict__ partials) {
  const int tid = threadIdx.x;
  const int gid = blockIdx.x * NTHR + tid;
  const int total = NBLK * NTHR;

  float acc_sq = 0.0f;
  float acc_pen = 0.0f;

  // Vectorized main loop: coalesced float4 (global_load_b128) streaming.
  const int n4 = n >> 2;
  const float4* p4 = reinterpret_cast<const float4*>(pred);
  const float4* t4 = reinterpret_cast<const float4*>(targ);
  for (int i = gid; i < n4; i += total) {
    float4 p = p4[i];
    float4 t = t4[i];
    float d0 = p.x - t.x, d1 = p.y - t.y, d2 = p.z - t.z, d3 = p.w - t.w;
    acc_sq = fmaf(d0, d0, acc_sq);
    acc_sq = fmaf(d1, d1, acc_sq);
    acc_sq = fmaf(d2, d2, acc_sq);
    acc_sq = fmaf(d3, d3, acc_sq);
    acc_pen += sign_mismatch(p.x, t.x);
    acc_pen += sign_mismatch(p.y, t.y);
    acc_pen += sign_mismatch(p.z, t.z);
    acc_pen += sign_mismatch(p.w, t.w);
  }
  // Scalar tail (n % 4), handled once by global thread 0.
  if (gid == 0) {
    for (int i = n4 << 2; i < n; ++i) {
      float d = pred[i] - targ[i];
      acc_sq = fmaf(d, d, acc_sq);
      acc_pen += sign_mismatch(pred[i], targ[i]);
    }
  }

  // Wave32 reduction via f32 WMMA (EXEC all-1s here: no divergence).
  float wsq = wave_sum_f32(acc_sq);
  float wpen = wave_sum_f32(acc_pen);

  __shared__ float red_sq[WAVES_PER_BLK];
  __shared__ float red_pen[WAVES_PER_BLK];
  const int lane = tid & 31;
  const int wave = tid >> 5;
  if (lane == 0) {
    red_sq[wave] = wsq;
    red_pen[wave] = wpen;
  }
  __syncthreads();
  if (tid == 0) {
    float bsq = 0.0f, bpen = 0.0f;
    for (int w = 0; w < WAVES_PER_BLK; ++w) {
      bsq += red_sq[w];
      bpen += red_pen[w];
    }
    partials[blockIdx.x] = make_float2(bsq, bpen);
  }
}

__global__ void mse_penalty_final_kernel(const float2* __restrict__ partials,
                                         int n, float* __restrict__ out) {
  const int tid = threadIdx.x;
  float acc_sq = 0.0f, acc_pen = 0.0f;
  for (int i = tid; i < NBLK; i += NTHR) {
    float2 v = partials[i];
    acc_sq += v.x;
    acc_pen += v.y;
  }
  float wsq = wave_sum_f32(acc_sq);
  float wpen = wave_sum_f32(acc_pen);

  __shared__ float red_sq[WAVES_PER_BLK];
  __shared__ float red_pen[WAVES_PER_BLK];
  const int lane = tid & 31;
  const int wave = tid >> 5;
  if (lane == 0) {
    red_sq[wave] = wsq;
    red_pen[wave] = wpen;
  }
  __syncthreads();
  if (tid == 0) {
    float tsq = 0.0f, tpen = 0.0f;
    for (int w = 0; w < WAVES_PER_BLK; ++w) {
      tsq += red_sq[w];
      tpen += red_pen[w];
    }
    // PENALTY_FACTOR == 1.0: result = mean(diff^2) + mean(mismatch)
    out[0] = (tsq + tpen) * (1.0f / (float)n);
  }
}

extern "C" void kernel_launch(void* const* d_in, const int* in_sizes, int n_in,
                              void* d_out, int out_size, void* d_ws, size_t ws_size,
                              hipStream_t stream) {
  const float* pred = (const float*)d_in[0];
  const float* targ = (const float*)d_in[1];
  float* out = (float*)d_out;
  float2* partials = (float2*)d_ws;  // NBLK * 8 bytes = 16 KB
  const int n = in_sizes[0];

  mse_penalty_partial_kernel<<<NBLK, NTHR, 0, stream>>>(pred, targ, n, partials);
  mse_penalty_final_kernel<<<1, NTHR, 0, stream>>>(partials, n, out);
}